// ThreshOtsu_85684597555371
// MI455X (gfx1250) — compile-verified
//
#include <hip/hip_runtime.h>
#include <hip/hip_bf16.h>

#define ROWS   24
#define NPE    (1u << 20)      // elements per row (1024*1024)
#define NQ4    (NPE / 4)       // float4 per row (2^18)
#define NBINS  256
#define MAXTRY 25              // int(256 * 0.1)
#define BPR    64              // blocks per row for streaming passes

typedef __attribute__((ext_vector_type(2))) float v2f;
typedef __attribute__((ext_vector_type(8))) float v8f;

// ---- order-preserving float <-> uint encoding for atomic min/max ----
__device__ __forceinline__ unsigned f2ord(float f) {
  unsigned u = __float_as_uint(f);
  return (u & 0x80000000u) ? ~u : (u | 0x80000000u);
}
__device__ __forceinline__ float ord2f(unsigned k) {
  unsigned u = (k & 0x80000000u) ? (k & 0x7FFFFFFFu) : ~k;
  return __uint_as_float(u);
}

// =====================================================================
// Kernel 0: init workspace
// =====================================================================
__global__ void otsu_init(unsigned* minenc, unsigned* maxenc,
                          unsigned* g_cnt, float* g_s1, float* g_s2) {
  int i = blockIdx.x * 256 + threadIdx.x;           // 24 blocks * 256 = 6144
  if (i < ROWS) { minenc[i] = 0xFFFFFFFFu; maxenc[i] = 0u; }
  if (i < ROWS * NBINS) { g_cnt[i] = 0u; g_s1[i] = 0.0f; g_s2[i] = 0.0f; }
}

// =====================================================================
// Kernel 1: per-row min / max (float4 streaming, LDS tree, atomic combine)
// =====================================================================
__global__ void otsu_minmax(const float* __restrict__ x,
                            unsigned* minenc, unsigned* maxenc) {
  int row = blockIdx.x >> 6, chunk = blockIdx.x & (BPR - 1);
  const float4* xr = (const float4*)(x + (size_t)row * NPE);
  int base = chunk * (NQ4 / BPR);                   // 4096 float4 per block
  float mn = INFINITY, mx = -INFINITY;
#pragma unroll 4
  for (int k = 0; k < 16; ++k) {
    float4 v = xr[base + threadIdx.x + k * 256];
    mn = fminf(mn, fminf(fminf(v.x, v.y), fminf(v.z, v.w)));
    mx = fmaxf(mx, fmaxf(fmaxf(v.x, v.y), fmaxf(v.z, v.w)));
  }
  __shared__ float smn[256], smx[256];
  smn[threadIdx.x] = mn; smx[threadIdx.x] = mx;
  __syncthreads();
  for (int s = 128; s > 0; s >>= 1) {
    if (threadIdx.x < s) {
      smn[threadIdx.x] = fminf(smn[threadIdx.x], smn[threadIdx.x + s]);
      smx[threadIdx.x] = fmaxf(smx[threadIdx.x], smx[threadIdx.x + s]);
    }
    __syncthreads();
  }
  if (threadIdx.x == 0) {
    atomicMin(&minenc[row], f2ord(smn[0]));
    atomicMax(&maxenc[row], f2ord(smx[0]));
  }
}

// =====================================================================
// Kernel 2: per-row 256-bin histogram of (count, sum xs, sum xs^2)
// =====================================================================
__global__ void otsu_hist(const float* __restrict__ x,
                          const unsigned* __restrict__ minenc,
                          const unsigned* __restrict__ maxenc,
                          unsigned* g_cnt, float* g_s1, float* g_s2) {
  __shared__ unsigned hc[NBINS];
  __shared__ float h1[NBINS], h2[NBINS];
  __shared__ float s_invw, s_rmin;
  int row = blockIdx.x >> 6, chunk = blockIdx.x & (BPR - 1);
  hc[threadIdx.x] = 0u; h1[threadIdx.x] = 0.0f; h2[threadIdx.x] = 0.0f;
  if (threadIdx.x == 0) {
    float g = 0.0f;
    for (int r = 0; r < ROWS; ++r)
      g = fmaxf(g, ord2f(maxenc[r]) - ord2f(minenc[r]));
    s_invw = (float)NBINS / g;
    s_rmin = ord2f(minenc[row]);
  }
  __syncthreads();
  float invw = s_invw, rmin = s_rmin;
  const float4* xr = (const float4*)(x + (size_t)row * NPE);
  int base = chunk * (NQ4 / BPR);
  for (int k = 0; k < 16; ++k) {
    int i4 = base + threadIdx.x + k * 256;
    if (k + 1 < 16) __builtin_prefetch(&xr[i4 + 256], 0, 0);   // global_prefetch_b8
    float4 v = xr[i4];
    float e[4] = {v.x, v.y, v.z, v.w};
#pragma unroll
    for (int j = 0; j < 4; ++j) {
      float xs = e[j] - rmin;
      int b = (int)floorf(xs * invw);
      b = min(max(b, 0), NBINS - 1);
      atomicAdd(&hc[b], 1u);        // ds_add_u32
      atomicAdd(&h1[b], xs);        // ds_add_f32
      atomicAdd(&h2[b], xs * xs);
    }
  }
  __syncthreads();
  int o = row * NBINS + threadIdx.x;
  atomicAdd(&g_cnt[o], hc[threadIdx.x]);
  atomicAdd(&g_s1[o], h1[threadIdx.x]);
  atomicAdd(&g_s2[o], h2[threadIdx.x]);
}

// =====================================================================
// WMMA segmented prefix: P = M(16x16) x U(16x16 upper-tri ones)
// via 4 chained v_wmma_f32_16x16x4_f32. src/dst are 256-float LDS arrays.
// =====================================================================
__device__ __forceinline__ void wmma_seg_prefix(const float* src, float* dst, int lane) {
  int m = lane & 15;         // A row / B,D column
  int hi = lane >> 4;        // lane half selects K (+2) for A/B, M (+8) for D
  v8f acc = {0.f, 0.f, 0.f, 0.f, 0.f, 0.f, 0.f, 0.f};
#pragma unroll
  for (int c = 0; c < 4; ++c) {
    int k0 = 4 * c + 2 * hi;
    v2f a, b;
    a.x = src[16 * m + k0];
    a.y = src[16 * m + k0 + 1];
    b.x = (k0     <= m) ? 1.0f : 0.0f;   // U[k][n] = (k <= n), n == m
    b.y = (k0 + 1 <= m) ? 1.0f : 0.0f;
    acc = __builtin_amdgcn_wmma_f32_16x16x4_f32(false, a, false, b,
                                                (short)0, acc, false, false);
  }
#pragma unroll
  for (int j = 0; j < 8; ++j)
    dst[16 * (j + 8 * hi) + m] = acc[j];   // D vgpr j -> rows j / j+8
}

// =====================================================================
// Kernel 3: single-block Otsu solve (cumsums via WMMA, icv, coupled scan)
// =====================================================================
__global__ void otsu_solve(const unsigned* __restrict__ minenc,
                           const unsigned* __restrict__ maxenc,
                           const unsigned* __restrict__ g_cnt,
                           const float* __restrict__ g_s1,
                           const float* __restrict__ g_s2,
                           float* __restrict__ g_thresh) {
  __shared__ float hsrc[3 * NBINS];
  __shared__ float cum[3 * NBINS];
  __shared__ float icv_s[ROWS * NBINS];
  __shared__ float s_rmin[ROWS];
  __shared__ float s_width;
  int tid = threadIdx.x;

  if (tid == 0) {
    float g = 0.0f;
    for (int r = 0; r < ROWS; ++r) {
      float mn = ord2f(minenc[r]), mx = ord2f(maxenc[r]);
      s_rmin[r] = mn;
      g = fmaxf(g, mx - mn);
    }
    s_width = g * (1.0f / (float)NBINS);
  }
  __syncthreads();

  for (int r = 0; r < ROWS; ++r) {
    int o = r * NBINS + tid;
    hsrc[tid]             = (float)g_cnt[o];
    hsrc[NBINS + tid]     = g_s1[o];
    hsrc[2 * NBINS + tid] = g_s2[o];
    __syncthreads();

    int wave = tid >> 5;
    if (wave < 3)                                        // 3 full waves, EXEC all-1s
      wmma_seg_prefix(&hsrc[wave * NBINS], &cum[wave * NBINS], tid & 31);
    __syncthreads();

    // add segment offsets: off[m] = sum_{k<m} P[k][15]
    float offreg = 0.0f;
    int q = tid >> 4, mm = tid & 15;
    if (tid < 48)
      for (int k2 = 0; k2 < mm; ++k2)
        offreg += cum[q * NBINS + 16 * k2 + 15];
    __syncthreads();
    if (tid < 48)
      for (int n = 0; n < 16; ++n)
        cum[q * NBINS + 16 * mm + n] += offreg;
    __syncthreads();

    // inter-class variance at threshold k = tid
    float Nf  = cum[NBINS - 1];
    float n0  = cum[tid];
    float S1p = cum[NBINS + tid],     S2p = cum[2 * NBINS + tid];
    float T1  = cum[2 * NBINS - 1],   T2  = cum[3 * NBINS - 1];
    float mean = T1 / Nf;
    float S1_0 = (n0 > 0.5f) ? (S1p - n0 * mean) : 0.0f;
    float S2_0 = (n0 > 0.5f) ? (S2p - 2.0f * mean * S1p + n0 * mean * mean) : 0.0f;
    float tot1 = T1 - Nf * mean;
    float tot2 = T2 - 2.0f * mean * T1 + Nf * mean * mean;
    float n1 = Nf - n0;
    float v0 = (S2_0 - S1_0 * S1_0 / fmaxf(n0, 1.0f)) / fmaxf(n0 - 1.0f, 1.0f);
    float S1_1 = tot1 - S1_0, S2_1 = tot2 - S2_0;
    float v1 = (S2_1 - S1_1 * S1_1 / fmaxf(n1, 1.0f)) / fmaxf(n1 - 1.0f, 1.0f);
    float w0 = n0 / Nf, w1 = 1.0f - w0;
    float d = v0 - v1;
    icv_s[r * NBINS + tid] = ((n0 > 1.5f) && (n1 > 1.5f)) ? (w0 * w1 * d * d) : -1.0f;
    __syncthreads();
  }

  // coupled early-stop argmax scan, one wave (lane r tracks row r)
  if (tid < 32) {
    int r = tid;
    bool act = r < ROWS;
    float maxicv = 0.0f, best = 0.0f;
    int no_upd = 0;
    bool stopped = false;
    float width = s_width;
    for (int b = 0; b < NBINS; ++b) {
      float v = act ? icv_s[r * NBINS + b] : -1.0f;
      bool upd = act && !stopped && (v > maxicv);
      if (upd) { maxicv = v; best = width * (float)(b + 1); }
      bool anyupd = __any((int)upd) != 0;
      no_upd = anyupd ? 0 : (no_upd + 1);
      stopped = stopped || (no_upd >= MAXTRY);
    }
    if (act) g_thresh[r] = best + s_rmin[r];
  }
}

// =====================================================================
// Kernel 4: apply threshold (float4 streaming)
// =====================================================================
__global__ void otsu_apply(const float* __restrict__ x,
                           const float* __restrict__ thr,
                           float* __restrict__ out) {
  const float4* x4 = (const float4*)x;
  float4* o4 = (float4*)out;
  const int total4 = ROWS * (int)NQ4;
  int stride = gridDim.x * 256;
  for (int i = blockIdx.x * 256 + threadIdx.x; i < total4; i += stride) {
    int r = i >> 18;                      // (i*4) / 2^20
    float th = thr[r];
    float4 v = x4[i];
    float4 o;
    o.x = (v.x < th) ? 0.0f : v.x;
    o.y = (v.y < th) ? 0.0f : v.y;
    o.z = (v.z < th) ? 0.0f : v.z;
    o.w = (v.w < th) ? 0.0f : v.w;
    o4[i] = o;
  }
}

extern "C" void kernel_launch(void* const* d_in, const int* in_sizes, int n_in,
                              void* d_out, int out_size, void* d_ws, size_t ws_size,
                              hipStream_t stream) {
  const float* x = (const float*)d_in[0];
  float* out = (float*)d_out;

  // workspace layout (74 KB)
  unsigned* minenc   = (unsigned*)d_ws;
  unsigned* maxenc   = minenc + ROWS;
  float*    g_thresh = (float*)(maxenc + ROWS);
  unsigned* g_cnt    = (unsigned*)(g_thresh + ROWS);
  float*    g_s1     = (float*)(g_cnt + ROWS * NBINS);
  float*    g_s2     = g_s1 + ROWS * NBINS;

  otsu_init  <<<ROWS, 256, 0, stream>>>(minenc, maxenc, g_cnt, g_s1, g_s2);
  otsu_minmax<<<ROWS * BPR, 256, 0, stream>>>(x, minenc, maxenc);
  otsu_hist  <<<ROWS * BPR, 256, 0, stream>>>(x, minenc, maxenc, g_cnt, g_s1, g_s2);
  otsu_solve <<<1, 256, 0, stream>>>(minenc, maxenc, g_cnt, g_s1, g_s2, g_thresh);
  otsu_apply <<<4096, 256, 0, stream>>>(x, g_thresh, out);
}